// Network_28587302323132
// MI455X (gfx1250) — compile-verified
//
#include <hip/hip_runtime.h>

typedef __attribute__((ext_vector_type(2))) float v2f;
typedef __attribute__((ext_vector_type(8))) float v8f;

#define GIN_D 512
#define BM 64
#define BN 64
#define BK 32
#define LDX 36  // Xs row stride (floats): 16B-aligned rows, conflict-free strided reads
#define LDK 34  // Wt row stride (floats): 8B-aligned b64 fragment reads, conflict-free

// out = in (vectorized copy of the self term before scatter-add)
__global__ void copy_f4(const float4* __restrict__ in, float4* __restrict__ out, long n4) {
    long i = (long)blockIdx.x * blockDim.x + threadIdx.x;
    if (i < n4) out[i] = in[i];
}

// agg[dst[e]] += x[src[e]] ; 128 threads per edge, 4 floats per thread.
// agg buffer (102.4 MB) fits in the 192 MB L2, so these f32 atomics resolve in L2.
__global__ void edge_scatter_add(const float* __restrict__ x, const int* __restrict__ src,
                                 const int* __restrict__ dst, float* __restrict__ agg,
                                 int n_edges) {
    long tid  = (long)blockIdx.x * blockDim.x + threadIdx.x;
    int edge  = (int)(tid >> 7);
    int lane  = (int)(tid & 127);
    if (edge >= n_edges) return;
    int s = src[edge], d = dst[edge];
    const float4 v = *(const float4*)(x + (size_t)s * GIN_D + lane * 4);
    float* p = agg + (size_t)d * GIN_D + lane * 4;
    atomicAdd(p + 0, v.x);
    atomicAdd(p + 1, v.y);
    atomicAdd(p + 2, v.z);
    atomicAdd(p + 3, v.w);
}

// Load one 64x32 X tile + 32x64 W tile worth of data into registers (prefetch).
__device__ __forceinline__ void stage_load(const float* __restrict__ X,
                                           const float* __restrict__ W, int nrows,
                                           int m0, int n0, int k0, int t,
                                           float4 rx[2], float4 rw[2]) {
#pragma unroll
    for (int i = 0; i < 2; i++) {
        const int q   = t + i * 256;
        const int row = q >> 3;       // X tile: 64 rows x 8 float4
        const int c4  = q & 7;
        const int gr  = m0 + row;
        rx[i] = make_float4(0.f, 0.f, 0.f, 0.f);
        if (gr < nrows) rx[i] = *(const float4*)(X + (size_t)gr * GIN_D + k0 + c4 * 4);
        const int wr  = q >> 4;       // W tile: 32 rows x 16 float4
        const int wc4 = q & 15;
        rw[i] = *(const float4*)(W + (size_t)(k0 + wr) * GIN_D + n0 + wc4 * 4);
    }
}

// Y = relu(X @ W + b), X:[nrows,512], W:[512,512] row-major, fp32 WMMA 16x16x4.
// 256 threads = 8 waves (4Mx2N); each wave owns 16x32 (two v8f accs, A reused).
// W tile stored K-interleaved (Wt[n][k]) so each B fragment is one ds_load_b64.
// Register-prefetch double buffering overlaps next-chunk global loads with WMMAs.
__global__ __launch_bounds__(256) void gemm_bias_relu(const float* __restrict__ X,
                                                      const float* __restrict__ W,
                                                      const float* __restrict__ bias,
                                                      float* __restrict__ Y, int nrows) {
    __shared__ __align__(16) float Xs[BM][LDX];
    __shared__ __align__(8)  float Wt[BN][LDK];  // transposed W tile: [n][k]

    const int t     = threadIdx.x;
    const int wave  = t >> 5;
    const int lane  = t & 31;
    const int waveM = wave & 3;   // 0..3 -> 16-row slice
    const int waveN = wave >> 2;  // 0..1 -> 32-col slice
    const int m0    = blockIdx.y * BM;
    const int n0    = blockIdx.x * BN;
    const int half  = lane >> 4;  // 0: lanes 0-15, 1: lanes 16-31
    const int l16   = lane & 15;

    v8f acc0 = {};
    v8f acc1 = {};

    float4 rx[2], rw[2];
    stage_load(X, W, nrows, m0, n0, 0, t, rx, rw);

    for (int k0 = 0; k0 < GIN_D; k0 += BK) {
        // Commit prefetched tile to LDS.
#pragma unroll
        for (int i = 0; i < 2; i++) {
            const int q   = t + i * 256;
            const int row = q >> 3;
            const int c4  = q & 7;
            *(float4*)&Xs[row][c4 * 4] = rx[i];
            const int wr  = q >> 4;
            const int wc4 = q & 15;
            Wt[wc4 * 4 + 0][wr] = rw[i].x;
            Wt[wc4 * 4 + 1][wr] = rw[i].y;
            Wt[wc4 * 4 + 2][wr] = rw[i].z;
            Wt[wc4 * 4 + 3][wr] = rw[i].w;
        }
        __syncthreads();

        // Prefetch next chunk while this chunk's WMMAs run.
        if (k0 + BK < GIN_D) stage_load(X, W, nrows, m0, n0, k0 + BK, t, rx, rw);

#pragma unroll
        for (int kk = 0; kk < BK; kk += 4) {
            // A 16x4 f32: lanes 0-15 hold K=kk,kk+1; lanes 16-31 hold K=kk+2,kk+3.
            // B 4x16 f32: symmetric; Wt layout makes (W[k][n],W[k+1][n]) contiguous.
            const int ka = kk + half * 2;
            v2f a  = *(const v2f*)&Xs[waveM * 16 + l16][ka];
            v2f b0 = *(const v2f*)&Wt[waveN * 32 + l16][ka];
            v2f b1 = *(const v2f*)&Wt[waveN * 32 + 16 + l16][ka];
            acc0 = __builtin_amdgcn_wmma_f32_16x16x4_f32(
                false, a, false, b0, (short)0, acc0, false, false);
            acc1 = __builtin_amdgcn_wmma_f32_16x16x4_f32(
                false, a, false, b1, (short)0, acc1, false, false);
        }
        __syncthreads();
    }

    // Epilogue: C/D layout -> VGPR i: M = i + 8*half, N = lane&15
    const int colg0 = n0 + waveN * 32 + l16;
    const int colg1 = colg0 + 16;
    const float bv0 = bias[colg0];
    const float bv1 = bias[colg1];
#pragma unroll
    for (int i = 0; i < 8; i++) {
        int rowg = m0 + waveM * 16 + half * 8 + i;
        if (rowg < nrows) {
            float v0 = acc0[i] + bv0;
            float v1 = acc1[i] + bv1;
            Y[(size_t)rowg * GIN_D + colg0] = v0 > 0.f ? v0 : 0.f;
            Y[(size_t)rowg * GIN_D + colg1] = v1 > 0.f ? v1 : 0.f;
        }
    }
}

__global__ void init_logits(float* __restrict__ out, const float* __restrict__ bc, int n) {
    int i = blockIdx.x * blockDim.x + threadIdx.x;
    if (i < n) out[i] = bc[0];
}

// logits[graph_ids[v]] += dot(h[v], Wc) ; one wave32 per node, shfl reduction.
__global__ void node_readout(const float* __restrict__ h, const float* __restrict__ Wc,
                             const int* __restrict__ gid, float* __restrict__ out,
                             int nnodes) {
    int node = blockIdx.x * 8 + (threadIdx.x >> 5);
    int lane = threadIdx.x & 31;
    if (node >= nnodes) return;
    const float* hp = h + (size_t)node * GIN_D;
    float p = 0.f;
#pragma unroll
    for (int i = 0; i < GIN_D / 32; i++) p += hp[lane + i * 32] * Wc[lane + i * 32];
#pragma unroll
    for (int off = 16; off > 0; off >>= 1) p += __shfl_xor(p, off, 32);
    if (lane == 0) atomicAdd(&out[gid[node]], p);
}

extern "C" void kernel_launch(void* const* d_in, const int* in_sizes, int n_in,
                              void* d_out, int out_size, void* d_ws, size_t ws_size,
                              hipStream_t stream) {
    const float* x   = (const float*)d_in[0];
    const int*   src = (const int*)d_in[1];
    const int*   dst = (const int*)d_in[2];
    const int*   gid = (const int*)d_in[3];
    // d_in[4] = num_graphs scalar (use out_size instead)
    const float* W1 = (const float*)d_in[5];
    const float* b1 = (const float*)d_in[6];
    const float* W2 = (const float*)d_in[7];
    const float* b2 = (const float*)d_in[8];
    const float* W3 = (const float*)d_in[9];
    const float* b3 = (const float*)d_in[10];
    const float* Wc = (const float*)d_in[11];
    const float* bc = (const float*)d_in[12];

    const int nnodes = in_sizes[0] / GIN_D;
    const int nedges = in_sizes[1];
    const size_t node_elems = (size_t)nnodes * GIN_D;

    float* buf0 = (float*)d_ws;          // agg buffer (self + neighbors)
    float* buf1 = buf0 + node_elems;     // GEMM output / next layer input

    const long n4 = (long)(node_elems / 4);
    dim3 cpGrid((unsigned)((n4 + 255) / 256));
    const long scThreads = (long)nedges * 128;
    dim3 scGrid((unsigned)((scThreads + 255) / 256));
    dim3 gmGrid(GIN_D / BN, (nnodes + BM - 1) / BM);

    const float* Wl[3] = {W1, W2, W3};
    const float* bl[3] = {b1, b2, b3};
    const float* cur = x;
    for (int l = 0; l < 3; l++) {
        copy_f4<<<cpGrid, 256, 0, stream>>>((const float4*)cur, (float4*)buf0, n4);
        edge_scatter_add<<<scGrid, 256, 0, stream>>>(cur, src, dst, buf0, nedges);
        gemm_bias_relu<<<gmGrid, 256, 0, stream>>>(buf0, Wl[l], bl[l], buf1, nnodes);
        cur = buf1;  // GEMM reads only buf0, so writing buf1 in-place across layers is safe
    }

    init_logits<<<(out_size + 255) / 256, 256, 0, stream>>>((float*)d_out, bc, out_size);
    node_readout<<<(nnodes + 7) / 8, 256, 0, stream>>>(buf1, Wc, gid, (float*)d_out, nnodes);
}